// Encoder_137438954180
// MI455X (gfx1250) — compile-verified
//
#include <hip/hip_runtime.h>
#include <math.h>

#define DD 128   // attrs_dim == hidden_dim

typedef float v2f __attribute__((ext_vector_type(2)));
typedef float v8f __attribute__((ext_vector_type(8)));

// ---------------------------------------------------------------------------
// Degree / normalization
// ---------------------------------------------------------------------------
__global__ void k_deg_init(float* __restrict__ deg, int n) {
  int i = blockIdx.x * blockDim.x + threadIdx.x;
  if (i < n) deg[i] = 1.0f;  // self-loop contribution
}

__global__ void k_deg_accum(const int* __restrict__ col, float* __restrict__ deg, int e) {
  int i = blockIdx.x * blockDim.x + threadIdx.x;
  if (i < e) atomicAdd(&deg[col[i]], 1.0f);
}

__global__ void k_deg_rsqrt(float* __restrict__ deg, int n) {
  int i = blockIdx.x * blockDim.x + threadIdx.x;
  if (i < n) {
    float d = deg[i];
    deg[i] = (d > 0.0f) ? rsqrtf(d) : 0.0f;
  }
}

// ---------------------------------------------------------------------------
// GEMM: H[N,128] = X[N,128] @ W[128,128] using V_WMMA_F32_16X16X4_F32
// Block = 256 threads = 8 waves; block covers 128 rows (16 rows / wave).
//
// W is staged in LDS *swizzled* so a B fragment is one aligned ds_load_b64:
//   Wswz[((k>>2)*128 + n)*4 + (k&3)] = W[k][n]
// For a given K-group k4 = k>>2, lane reads the pair (khalf, khalf+1) of the
// 4-float cell for its column -> contiguous 8 bytes.
//
// Fragment layouts per CDNA5 ISA 7.12.2:
//  A 16x4 : lanes 0-15 -> (M=lane, K={0,1}); lanes 16-31 -> K={2,3}
//  B 4x16 : v0 lanes0-15 K=0, lanes16-31 K=2 ; v1 lanes0-15 K=1, lanes16-31 K=3
//  C 16x16: VGPR v, lanes0-15 M=v, lanes16-31 M=v+8, N=lane%16
// ---------------------------------------------------------------------------
__global__ __launch_bounds__(256) void k_gemm_xw(const float* __restrict__ X,
                                                 const float* __restrict__ W,
                                                 float* __restrict__ H,
                                                 int n_rows) {
  __shared__ float Wl[DD * DD];  // 64 KB swizzled (WGP has 320 KB LDS)
  const int tid = threadIdx.x;
  // cooperative swizzled stage: each thread moves float4 chunks of W rows
  for (int idx = tid; idx < DD * DD / 4; idx += 256) {
    const int k = idx >> 5;          // row of W
    const int n = (idx & 31) << 2;   // starting column (multiple of 4)
    const float4 w = *reinterpret_cast<const float4*>(&W[k * DD + n]);
    const int base = (k >> 2) * DD * 4 + (k & 3);
    Wl[base + (n + 0) * 4] = w.x;
    Wl[base + (n + 1) * 4] = w.y;
    Wl[base + (n + 2) * 4] = w.z;
    Wl[base + (n + 3) * 4] = w.w;
  }
  __syncthreads();

  const int wave  = tid >> 5;
  const int lane  = tid & 31;
  const int mbase = blockIdx.x * 128 + wave * 16;
  int mrow = mbase + (lane & 15);
  if (mrow >= n_rows) mrow = n_rows - 1;  // clamp (dup reads OK; store guarded)
  const int khalf = (lane >> 4) << 1;     // 0 for lanes 0-15, 2 for lanes 16-31
  const int ncol  = lane & 15;

  v8f zero = {};
  v8f acc[8];
#pragma unroll
  for (int t = 0; t < 8; ++t) acc[t] = zero;

  const float* xrow = X + (size_t)mrow * DD;
  for (int k = 0; k < DD; k += 4) {
    // A fragment: contiguous pair -> single global_load_b64
    v2f a = *reinterpret_cast<const v2f*>(&xrow[k + khalf]);
    const int kcell = (k >> 2) * DD * 4 + khalf;
#pragma unroll
    for (int t = 0; t < 8; ++t) {
      // B fragment: contiguous pair in swizzled LDS -> single ds_load_b64
      v2f b = *reinterpret_cast<const v2f*>(&Wl[kcell + (t * 16 + ncol) * 4]);
      acc[t] = __builtin_amdgcn_wmma_f32_16x16x4_f32(
          false, a, false, b, (short)0, acc[t], false, false);
    }
  }

  const int moff = (lane >> 4) * 8;
#pragma unroll
  for (int t = 0; t < 8; ++t) {
#pragma unroll
    for (int v = 0; v < 8; ++v) {
      const int m = mbase + moff + v;
      if (m < n_rows) H[(size_t)m * DD + t * 16 + ncol] = acc[t][v];
    }
  }
}

// ---------------------------------------------------------------------------
// Self-loop seed: Acc[i,:] = H[i,:] * dinv[i]^2   (float4 over N*D)
// ---------------------------------------------------------------------------
__global__ void k_seed_self(const float* __restrict__ H, const float* __restrict__ dinv,
                            float* __restrict__ Acc, int total4) {
  int i = blockIdx.x * blockDim.x + threadIdx.x;
  if (i >= total4) return;
  int node = i >> 5;  // (i*4)/128
  float s = dinv[node];
  s = s * s;
  float4 v = reinterpret_cast<const float4*>(H)[i];
  v.x *= s; v.y *= s; v.z *= s; v.w *= s;
  reinterpret_cast<float4*>(Acc)[i] = v;
}

// ---------------------------------------------------------------------------
// Edge scatter: one wave per edge iteration; each lane moves 4 contiguous
// floats. Grid-stride so each wave handles ~8 edges and can prefetch the next
// gather row (global_prefetch_b8) while the current edge's atomics drain.
// Acc[col] += H[row] * (dinv[row]*dinv[col])
// ---------------------------------------------------------------------------
__global__ __launch_bounds__(256) void k_edge_scatter(const float* __restrict__ H,
                                                      const int* __restrict__ row,
                                                      const int* __restrict__ col,
                                                      const float* __restrict__ dinv,
                                                      float* __restrict__ Acc,
                                                      int n_edges) {
  const int lane    = threadIdx.x & 31;
  const int wstride = (int)(gridDim.x * 8);
  for (int e = (int)((blockIdx.x * 256u + threadIdx.x) >> 5); e < n_edges; e += wstride) {
    const int en = e + wstride;
    if (en < n_edges) {
      // prefetch next iteration's gather row into the cache hierarchy
      __builtin_prefetch(&H[(size_t)row[en] * DD + lane * 4], 0, 3);
    }
    const int r = row[e];
    const int c = col[e];
    const float nrm = dinv[r] * dinv[c];
    const float4 hv = *reinterpret_cast<const float4*>(&H[(size_t)r * DD + lane * 4]);
    float* o = &Acc[(size_t)c * DD + lane * 4];
    atomicAdd(o + 0, hv.x * nrm);
    atomicAdd(o + 1, hv.y * nrm);
    atomicAdd(o + 2, hv.z * nrm);
    atomicAdd(o + 3, hv.w * nrm);
  }
}

// ---------------------------------------------------------------------------
// Finalize: Acc = tanh(Acc + b)   (float4 over N*D)
// ---------------------------------------------------------------------------
__global__ void k_finalize_tanh(float* __restrict__ Acc, const float* __restrict__ b,
                                int total4) {
  int i = blockIdx.x * blockDim.x + threadIdx.x;
  if (i >= total4) return;
  int dbase = (i & 31) * 4;  // (i*4) % 128
  float4 v = reinterpret_cast<float4*>(Acc)[i];
  v.x = tanhf(v.x + b[dbase + 0]);
  v.y = tanhf(v.y + b[dbase + 1]);
  v.z = tanhf(v.z + b[dbase + 2]);
  v.w = tanhf(v.w + b[dbase + 3]);
  reinterpret_cast<float4*>(Acc)[i] = v;
}

// ---------------------------------------------------------------------------
// Pooling helpers
// ---------------------------------------------------------------------------
__global__ void k_zero(float* __restrict__ p, int n) {
  int i = blockIdx.x * blockDim.x + threadIdx.x;
  if (i < n) p[i] = 0.0f;
}

__global__ void k_pool_count(const int* __restrict__ batch, float* __restrict__ cnt, int n) {
  int i = blockIdx.x * blockDim.x + threadIdx.x;
  if (i < n) atomicAdd(&cnt[batch[i]], 1.0f);
}

__global__ void k_pool_sum(const float* __restrict__ x, const int* __restrict__ batch,
                           float* __restrict__ xmean, int total) {
  int i = blockIdx.x * blockDim.x + threadIdx.x;
  if (i >= total) return;
  int node = i >> 7;       // i / 128
  int d = i & (DD - 1);    // i % 128
  atomicAdd(&xmean[(size_t)batch[node] * DD + d], x[i]);
}

__global__ void k_pool_div(float* __restrict__ xmean, const float* __restrict__ cnt,
                           int total) {
  int i = blockIdx.x * blockDim.x + threadIdx.x;
  if (i >= total) return;
  xmean[i] /= fmaxf(cnt[i >> 7], 1.0f);
}

__global__ void k_copy4(const float* __restrict__ src, float* __restrict__ dst, int total4) {
  int i = blockIdx.x * blockDim.x + threadIdx.x;
  if (i < total4)
    reinterpret_cast<float4*>(dst)[i] = reinterpret_cast<const float4*>(src)[i];
}

// ---------------------------------------------------------------------------
// Launch
// ---------------------------------------------------------------------------
extern "C" void kernel_launch(void* const* d_in, const int* in_sizes, int n_in,
                              void* d_out, int out_size, void* d_ws, size_t ws_size,
                              hipStream_t stream) {
  const float* attrs      = (const float*)d_in[0];
  const int*   edge_index = (const int*)d_in[1];
  const int*   batch      = (const int*)d_in[2];
  const float* W[3] = {(const float*)d_in[3], (const float*)d_in[5], (const float*)d_in[7]};
  const float* B[3] = {(const float*)d_in[4], (const float*)d_in[6], (const float*)d_in[8]};

  const int Nn = in_sizes[0] / DD;                       // 50000
  const int Ee = in_sizes[1] / 2;                        // 800000
  const int Gg = (out_size - 4 * Nn * DD) / DD;          // 512

  const int* row = edge_index;        // sources
  const int* col = edge_index + Ee;   // targets

  // workspace layout
  float* h    = (float*)d_ws;                 // N*D
  float* dinv = h + (size_t)Nn * DD;          // N
  float* cnt  = dinv + Nn;                    // G

  // output layout: x | x_mean | xs0 | xs1 | xs2
  float* out   = (float*)d_out;
  float* x_out = out;
  float* xmean = out + (size_t)Nn * DD;
  float* xs[3];
  xs[0] = xmean + (size_t)Gg * DD;
  xs[1] = xs[0] + (size_t)Nn * DD;
  xs[2] = xs[1] + (size_t)Nn * DD;

  const int T = 256;
  const int nd4 = Nn * DD / 4;

  // 1) degrees -> dinv
  k_deg_init<<<(Nn + T - 1) / T, T, 0, stream>>>(dinv, Nn);
  k_deg_accum<<<(Ee + T - 1) / T, T, 0, stream>>>(col, dinv, Ee);
  k_deg_rsqrt<<<(Nn + T - 1) / T, T, 0, stream>>>(dinv, Nn);

  // edge-scatter grid: ~8 edges per wave so the prefetch lookahead is useful
  const int scat_blocks = (Ee + 63) / 64;

  // 2) three GCN layers
  const float* xin = attrs;
  for (int layer = 0; layer < 3; ++layer) {
    k_gemm_xw<<<(Nn + 127) / 128, 256, 0, stream>>>(xin, W[layer], h, Nn);
    k_seed_self<<<(nd4 + T - 1) / T, T, 0, stream>>>(h, dinv, xs[layer], nd4);
    k_edge_scatter<<<scat_blocks, 256, 0, stream>>>(h, row, col, dinv, xs[layer], Ee);
    k_finalize_tanh<<<(nd4 + T - 1) / T, T, 0, stream>>>(xs[layer], B[layer], nd4);
    xin = xs[layer];
  }

  // 3) x output = xs2
  k_copy4<<<(nd4 + T - 1) / T, T, 0, stream>>>(xs[2], x_out, nd4);

  // 4) scatter-mean pooling
  const int gd = Gg * DD;
  k_zero<<<(gd + T - 1) / T, T, 0, stream>>>(xmean, gd);
  k_zero<<<(Gg + T - 1) / T, T, 0, stream>>>(cnt, Gg);
  k_pool_count<<<(Nn + T - 1) / T, T, 0, stream>>>(batch, cnt, Nn);
  k_pool_sum<<<(Nn * DD + T - 1) / T, T, 0, stream>>>(xs[2], batch, xmean, Nn * DD);
  k_pool_div<<<(gd + T - 1) / T, T, 0, stream>>>(xmean, cnt, gd);
}